// Decoder_9139690405992
// MI455X (gfx1250) — compile-verified
//
#include <hip/hip_runtime.h>

// Problem constants (fixed by reference setup_inputs)
#define WORKERS 2048
#define TASKS   2048
#define ETYPE   16
#define ABIL    64

typedef __attribute__((ext_vector_type(2))) float v2f;
typedef __attribute__((ext_vector_type(4))) float v4f;
typedef __attribute__((ext_vector_type(8))) float v8f;

// ---------------------------------------------------------------------------
// Kernel 1: y = X @ W (2048x64 @ 64x1) via V_WMMA_F32_16X16X4_F32, then
//   p1 = sigmoid(y + b); p2 = (1-p1)/15; store (lr, lp2) = (log2(p1/p2), log2 p2)
// One wave handles 16 worker rows. B-matrix columns are all identical (= W),
// so every column of D holds the dot product; 16 WMMAs accumulate over K=64.
// A 16x4 f32 layout (ISA 7.12.2): lane L holds row M=L%16, K = vgpr + 2*(L>=16).
// B 4x16 mirrors the same K-to-(lane-half,vgpr) mapping with rows striped
// across lanes, so per-lane B values depend only on K -> broadcast W works.
// ---------------------------------------------------------------------------
__global__ __launch_bounds__(32) void worker_logit_kernel(
    const float* __restrict__ inputs, const float* __restrict__ W,
    const float* __restrict__ bptr, float* __restrict__ wlp)
{
    const int lane = threadIdx.x;            // 0..31, one full wave32
    const int half = (lane >= 16) ? 1 : 0;   // which K pair this half-wave owns
    const int m0   = blockIdx.x * 16;
    const int row  = m0 + (lane & 15);
    const float* __restrict__ xrow = inputs + (size_t)row * ABIL;

    v8f c = {};
#pragma unroll
    for (int k = 0; k < ABIL; k += 4) {
        const int kk = k + 2 * half;
        v2f a;  a.x  = xrow[kk]; a.y  = xrow[kk + 1];
        v2f bm; bm.x = W[kk];    bm.y = W[kk + 1];
        // D = A*B + C ; every column of D equals y[m]
        c = __builtin_amdgcn_wmma_f32_16x16x4_f32(
                /*neg_a=*/false, a, /*neg_b=*/false, bm,
                /*c_mod=*/(short)0, c, /*reuse_a=*/false, /*reuse_b=*/false);
    }

    // C/D layout: VGPR r, lanes 0-15 -> M=r, lanes 16-31 -> M=r+8.
    // Lanes 0 and 16 extract rows 0..7 and 8..15 respectively.
    if ((lane & 15) == 0) {
        const float bb = bptr[0];
#pragma unroll
        for (int r = 0; r < 8; ++r) {
            const int m = m0 + r + 8 * half;
            const float x  = c[r] + bb;
            const float e  = __builtin_amdgcn_exp2f(-1.44269504f * x); // e^-x
            float p1 = __builtin_amdgcn_rcpf(1.0f + e);                // sigmoid
            p1 = fminf(p1, 0.99999994f);            // avoid log2(0) for p2
            const float p2  = (1.0f - p1) * (1.0f / 15.0f);
            const float lp1 = __builtin_amdgcn_logf(p1);               // log2
            const float lp2 = __builtin_amdgcn_logf(p2);
            wlp[2 * m]     = lp1 - lp2;   // lr
            wlp[2 * m + 1] = lp2;
        }
    }
}

// ---------------------------------------------------------------------------
// Kernel 2: P[w,t,e] = exp2(tau[t,e] * lr[w] + lp2[w])   (67.1M f32, 256 MB)
// Block = 256 threads: thread owns one (task, e-quad) -> tau float4 kept in
// registers and reused across a 32-worker loop. Writes are consecutive
// 16-byte chunks across the block -> perfectly coalesced 128-bit
// non-temporal stores (output is write-once; keep it out of L2/WGP$).
// ---------------------------------------------------------------------------
__global__ __launch_bounds__(256) void pmat_kernel(
    const float* __restrict__ inputs, const float* __restrict__ wlp,
    float* __restrict__ out)
{
    const int tid = threadIdx.x;
    const int t   = blockIdx.x * 64 + (tid >> 2);   // 64 tasks per block
    const int e4  = (tid & 3) << 2;                 // e in {0,4,8,12}

    // tau row for task t lives at inputs[(WORKERS + t) * ABIL + e], e < 16
    const v4f tau = *reinterpret_cast<const v4f*>(
        inputs + (size_t)(WORKERS + t) * ABIL + e4);

    const int w0 = blockIdx.y * 32;                 // 32 workers per block
    float* outp = out + (size_t)w0 * ((size_t)TASKS * ETYPE)
                      + (size_t)t * ETYPE + e4;

#pragma unroll 4
    for (int i = 0; i < 32; ++i) {
        const int w = w0 + i;
        const float lr  = wlp[2 * w];       // uniform across wave -> scalarized
        const float lp2 = wlp[2 * w + 1];
        v4f r;
        r.x = __builtin_amdgcn_exp2f(fmaf(tau.x, lr, lp2));
        r.y = __builtin_amdgcn_exp2f(fmaf(tau.y, lr, lp2));
        r.z = __builtin_amdgcn_exp2f(fmaf(tau.z, lr, lp2));
        r.w = __builtin_amdgcn_exp2f(fmaf(tau.w, lr, lp2));
        __builtin_nontemporal_store(r, reinterpret_cast<v4f*>(outp));
        outp += (size_t)TASKS * ETYPE;
    }
}

extern "C" void kernel_launch(void* const* d_in, const int* in_sizes, int n_in,
                              void* d_out, int out_size, void* d_ws, size_t ws_size,
                              hipStream_t stream)
{
    const float* inputs = (const float*)d_in[0];  // (4096, 64) f32
    const float* W      = (const float*)d_in[1];  // (64, 1)   f32
    const float* b      = (const float*)d_in[2];  // (1,)      f32
    float*       out    = (float*)d_out;          // (2048, 2048, 16) f32
    float*       wlp    = (float*)d_ws;           // 2048 * {lr, lp2} = 16 KB

    // Phase 1: per-worker logits -> (lr, lp2), WMMA GEMV. 128 waves.
    worker_logit_kernel<<<dim3(WORKERS / 16), dim3(32), 0, stream>>>(
        inputs, W, b, wlp);

    // Phase 2: streaming 256 MB output, bandwidth-bound (~11 us floor).
    pmat_kernel<<<dim3(TASKS / 64, WORKERS / 32), dim3(256), 0, stream>>>(
        inputs, wlp, out);
}